// ResidualVQLayer_6640019440246
// MI455X (gfx1250) — compile-verified
//
#include <hip/hip_runtime.h>
#include <hip/hip_bf16.h>

// VQ-VAE vector quantization, fused for MI455X (gfx1250, wave32).
//
// Dominant cost: 8192x8192x512 fp32 GEMM (z @ W^T) fused with row argmin.
// V_WMMA_F32_16X16X4_F32 keeps exact fp32 numerics (the discriminating
// cross-term is ~1e-3 against a 512 baseline -> bf16 would corrupt argmin).
// Whole working set (32 MB) is L2-resident (192 MB); W streams from L2.
// Arithmetic-intensity tuning: each wave computes TWO 16x16 C tiles sharing
// one B fragment (halves L2 traffic vs one tile), z tiles sit in LDS with a
// bank-conflict-free pad, and ||w||^2 is precomputed so the inner loop is
// pure loads + WMMA.

typedef float v2f __attribute__((ext_vector_type(2)));
typedef float v8f __attribute__((ext_vector_type(8)));

#define N_ROWS   8192      // 4 * 2048 latent vectors
#define K_CODES  8192      // codebook size
#define D_LAT    512       // latent dim
#define LDS_STRIDE 516     // 512 + 4 dword pad: lanes map to distinct banks
#define M_TILE   32        // rows per workgroup (2 x 16 sub-tiles per wave)

// ---------------------------------------------------------------------------
// Kernel 0: wnorm[n] = ||W[n]||^2, one wave per codebook row.
// ---------------------------------------------------------------------------
__launch_bounds__(256, 1)
__global__ void vq_wnorm_kernel(const float* __restrict__ W,
                                float* __restrict__ wnorm) {
  const int wave = threadIdx.x >> 5;
  const int lane = threadIdx.x & 31;
  const int row  = blockIdx.x * 8 + wave;
  const float* wrow = W + (size_t)row * D_LAT;
  float s = 0.f;
  #pragma unroll
  for (int j = 0; j < 4; ++j) {
    const float4 v = *reinterpret_cast<const float4*>(wrow + j * 128 + lane * 4);
    s += v.x * v.x + v.y * v.y + v.z * v.z + v.w * v.w;
  }
  #pragma unroll
  for (int off = 16; off >= 1; off >>= 1) s += __shfl_xor(s, off, 32);
  if (lane == 0) wnorm[row] = s;
}

// ---------------------------------------------------------------------------
// Kernel 1: fused distance GEMM (f32 WMMA) + per-row argmin over the codebook.
// Grid: 256 blocks x 128 threads (4 waves). Block owns 32 rows of z; each wave
// sweeps a disjoint 1/4 of the codebook columns in 16-wide tiles, computing
// two M sub-tiles per B fragment. argmin target: ||w||^2 - 2 z.w
// ---------------------------------------------------------------------------
__launch_bounds__(128, 1)
__global__ void vq_argmin_kernel(const float* __restrict__ z,
                                 const float* __restrict__ W,
                                 const float* __restrict__ wnorm,
                                 int*   __restrict__ idx_out,
                                 float* __restrict__ idx_f_out) {
  __shared__ float zlds[M_TILE * LDS_STRIDE];   // 66 KB of 320 KB WGP LDS
  __shared__ float red_v[4 * M_TILE];
  __shared__ int   red_n[4 * M_TILE];

  const int tid     = threadIdx.x;
  const int wave    = tid >> 5;
  const int lane    = tid & 31;
  const int laneRow = lane & 15;   // A-frag row / B-frag column / C-frag N
  const int laneHi  = lane >> 4;   // which K-pair half this lane holds
  const int rowBase = blockIdx.x * M_TILE;

  // Stage the 32x512 z tile into LDS: 128 threads x 32 float4 (coalesced).
  #pragma unroll
  for (int i = 0; i < 32; ++i) {
    const int e4 = tid + 128 * i;            // float4 index in tile (4096 total)
    const int r  = e4 >> 7;                  // 128 float4 per row
    const int c  = (e4 & 127) << 2;
    const float4 v =
        *reinterpret_cast<const float4*>(z + (size_t)(rowBase + r) * D_LAT + c);
    *reinterpret_cast<float4*>(&zlds[r * LDS_STRIDE + c]) = v;
  }
  __syncthreads();

  // Per-lane running argmin for both M sub-tiles. In the 16x16 f32 C/D layout
  // every element a lane holds shares column N = lane&15; VGPR i maps to row i
  // (lanes 0-15) or row i+8 (lanes 16-31).
  float bestv0[8], bestv1[8];
  int   bestn0[8], bestn1[8];
  #pragma unroll
  for (int i = 0; i < 8; ++i) {
    bestv0[i] = 3.4e38f; bestn0[i] = 0;
    bestv1[i] = 3.4e38f; bestn1[i] = 0;
  }

  for (int n0 = wave * 16; n0 < K_CODES; n0 += 64) {
    const int    col  = n0 + laneRow;
    const float* wrow = W + (size_t)col * D_LAT;
    const float  wn   = wnorm[col];          // per-lane scalar, hoisted

    v8f acc0 = {0.f, 0.f, 0.f, 0.f, 0.f, 0.f, 0.f, 0.f};
    v8f acc1 = acc0;

    // K=512 as 128 chained V_WMMA_F32_16X16X4_F32 per sub-tile; one B fragment
    // feeds two WMMAs (2:1 WMMA : global-load ratio).
    // A frag (16x4): lane holds z[M=lane&15][kk+2*hi], +1      (LDS, b64)
    // B frag (4x16): lane holds W[col][kk+2*hi], +1            (L2,  b64)
    #pragma unroll 4
    for (int kk = 0; kk < D_LAT; kk += 4) {
      const int ko = kk + 2 * laneHi;
      v2f b  = *reinterpret_cast<const v2f*>(wrow + ko);
      v2f a0 = *reinterpret_cast<const v2f*>(&zlds[laneRow * LDS_STRIDE + ko]);
      v2f a1 = *reinterpret_cast<const v2f*>(&zlds[(laneRow + 16) * LDS_STRIDE + ko]);
      acc0 = __builtin_amdgcn_wmma_f32_16x16x4_f32(
          false, a0, false, b, (short)0, acc0, false, false);
      acc1 = __builtin_amdgcn_wmma_f32_16x16x4_f32(
          false, a1, false, b, (short)0, acc1, false, false);
    }

    #pragma unroll
    for (int i = 0; i < 8; ++i) {
      const float d0 = wn - 2.0f * acc0[i];
      const float d1 = wn - 2.0f * acc1[i];
      if (d0 < bestv0[i]) { bestv0[i] = d0; bestn0[i] = col; }  // strict < :
      if (d1 < bestv1[i]) { bestv1[i] = d1; bestn1[i] = col; }  // first min
    }
  }

  // Reduce each row over the 16 lanes of its half-wave (xor stays in-half).
  #pragma unroll
  for (int i = 0; i < 8; ++i) {
    #pragma unroll
    for (int off = 8; off >= 1; off >>= 1) {
      float ov; int on;
      ov = __shfl_xor(bestv0[i], off, 32);
      on = __shfl_xor(bestn0[i], off, 32);
      if (ov < bestv0[i] || (ov == bestv0[i] && on < bestn0[i])) {
        bestv0[i] = ov; bestn0[i] = on;
      }
      ov = __shfl_xor(bestv1[i], off, 32);
      on = __shfl_xor(bestn1[i], off, 32);
      if (ov < bestv1[i] || (ov == bestv1[i] && on < bestn1[i])) {
        bestv1[i] = ov; bestn1[i] = on;
      }
    }
  }
  if (laneRow == 0) {  // lane 0 -> rows i, lane 16 -> rows i+8 (per sub-tile)
    #pragma unroll
    for (int i = 0; i < 8; ++i) {
      red_v[wave * M_TILE + laneHi * 8 + i]      = bestv0[i];
      red_n[wave * M_TILE + laneHi * 8 + i]      = bestn0[i];
      red_v[wave * M_TILE + 16 + laneHi * 8 + i] = bestv1[i];
      red_n[wave * M_TILE + 16 + laneHi * 8 + i] = bestn1[i];
    }
  }
  __syncthreads();

  // Cross-wave reduction (4 candidates per row), deterministic order.
  if (tid < M_TILE) {
    float bv = red_v[tid];
    int   bn = red_n[tid];
    #pragma unroll
    for (int w = 1; w < 4; ++w) {
      const float v = red_v[w * M_TILE + tid];
      const int   n = red_n[w * M_TILE + tid];
      if (v < bv || (v == bv && n < bn)) { bv = v; bn = n; }
    }
    const int row  = rowBase + tid;
    idx_out[row]   = bn;
    idx_f_out[row] = (float)bn;
  }
}

// ---------------------------------------------------------------------------
// Kernel 2: gather z_q = W[idx] (== straight-through "quantized" output) and
// per-row squared error. One wave per row; fixed shuffle tree (deterministic).
// ---------------------------------------------------------------------------
__launch_bounds__(256, 1)
__global__ void vq_gather_kernel(const float* __restrict__ z,
                                 const float* __restrict__ W,
                                 const int*   __restrict__ idx,
                                 float* __restrict__ qout,
                                 float* __restrict__ rowloss) {
  const int wave = threadIdx.x >> 5;
  const int lane = threadIdx.x & 31;
  const int row  = blockIdx.x * 8 + wave;

  const int    code = idx[row];                       // wave-uniform
  const float* wrow = W + (size_t)code * D_LAT;
  const float* zrow = z + (size_t)row  * D_LAT;
  float*       orow = qout + (size_t)row * D_LAT;

  float s = 0.f;
  #pragma unroll
  for (int j = 0; j < 4; ++j) {
    const int c = j * 128 + lane * 4;                 // coalesced b128
    const float4 q  = *reinterpret_cast<const float4*>(wrow + c);
    const float4 zv = *reinterpret_cast<const float4*>(zrow + c);
    *reinterpret_cast<float4*>(orow + c) = q;
    const float dx = zv.x - q.x, dy = zv.y - q.y;
    const float dz = zv.z - q.z, dw = zv.w - q.w;
    s += dx * dx + dy * dy + dz * dz + dw * dw;
  }
  #pragma unroll
  for (int off = 16; off >= 1; off >>= 1) s += __shfl_xor(s, off, 32);
  if (lane == 0) rowloss[row] = s;
}

// ---------------------------------------------------------------------------
// Kernel 3: single workgroup. Histogram in LDS (integer LDS atomics, order-
// independent), entropy + loss sums via fixed LDS trees -> deterministic.
// ---------------------------------------------------------------------------
__launch_bounds__(256, 1)
__global__ void vq_finalize_kernel(const int*   __restrict__ idx,
                                   const float* __restrict__ rowloss,
                                   float* __restrict__ loss_out,
                                   float* __restrict__ perp_out) {
  __shared__ unsigned hist[K_CODES];       // 32 KB of the 320 KB WGP LDS
  __shared__ float    redA[256];
  __shared__ float    redB[256];
  const int tid = threadIdx.x;

  for (int i = tid; i < K_CODES; i += 256) hist[i] = 0u;
  __syncthreads();
  for (int i = tid; i < N_ROWS; i += 256) atomicAdd(&hist[idx[i]], 1u);
  __syncthreads();

  float ls = 0.f, es = 0.f;
  for (int i = tid; i < K_CODES; i += 256) {   // N_ROWS == K_CODES == 8192
    ls += rowloss[i];
    const float p = (float)hist[i] * (1.0f / (float)N_ROWS);
    es += p * logf(p + 1e-10f);
  }
  redA[tid] = ls;
  redB[tid] = es;
  __syncthreads();
  for (int off = 128; off >= 1; off >>= 1) {
    if (tid < off) { redA[tid] += redA[tid + off]; redB[tid] += redB[tid + off]; }
    __syncthreads();
  }
  if (tid == 0) {
    *loss_out = 0.25f * redA[0] / (float)((size_t)N_ROWS * D_LAT);
    *perp_out = expf(-redB[0]);
  }
}

// ---------------------------------------------------------------------------
// d_out layout (floats): [quantized 4194304 | indices 8192 | loss | perplexity]
// d_ws: [int idx[8192] @0 | float rowloss[8192] @32768 | float wnorm[8192]
//        @65536] — fully rewritten every call (no cross-call state).
// ---------------------------------------------------------------------------
extern "C" void kernel_launch(void* const* d_in, const int* in_sizes, int n_in,
                              void* d_out, int out_size, void* d_ws, size_t ws_size,
                              hipStream_t stream) {
  const float* z = (const float*)d_in[0];   // [4, 2048, 512] f32
  const float* W = (const float*)d_in[1];   // [8192, 512]    f32

  float* out    = (float*)d_out;
  float* q_out  = out;                       // 4,194,304
  float* idx_f  = out + 4194304;             // 8,192
  float* loss   = out + 4202496;             // 1
  float* perp   = out + 4202497;             // 1

  int*   ws_idx     = (int*)d_ws;
  float* ws_rowloss = (float*)((char*)d_ws + 32768);
  float* ws_wnorm   = (float*)((char*)d_ws + 65536);

  vq_wnorm_kernel<<<K_CODES / 8, 256, 0, stream>>>(W, ws_wnorm);
  vq_argmin_kernel<<<N_ROWS / M_TILE, 128, 0, stream>>>(z, W, ws_wnorm,
                                                        ws_idx, idx_f);
  vq_gather_kernel<<<N_ROWS / 8, 256, 0, stream>>>(z, W, ws_idx, q_out,
                                                   ws_rowloss);
  vq_finalize_kernel<<<1, 256, 0, stream>>>(ws_idx, ws_rowloss, loss, perp);
}